// EPOT_49830210568883
// MI455X (gfx1250) — compile-verified
//
#include <hip/hip_runtime.h>
#include <math.h>

// ---------------------------------------------------------------------------
// EPOT (entropic partial Wasserstein) for MI455X / gfx1250.
// K[i,j] = exp(-10*C[i,j]) * R_i * Cc_j * S  -- only vectors + scalar evolve.
// C (16MB) stays L2-resident; exp recomputed on the fly (bandwidth-bound).
// Row GEMV uses V_WMMA_F32_16X16X4_F32 (fp32 precision preserved).
// ---------------------------------------------------------------------------

#define NEG_INV_EPS (-10.0f)   // -1/0.1
#define TOL_C        0.01f
#define S_MASS_C     0.9f
#define N_ITERS_C    50

constexpr int N  = 2048;
constexpr int M  = 2048;
constexpr int NM = N * M;
constexpr int STRIDE_ELEMS = 1024 * 256;   // threads in the big reduction grids

typedef __attribute__((ext_vector_type(2))) float v2f;
typedef __attribute__((ext_vector_type(8))) float v8f;

// Workspace layout (float offsets)
enum {
  OFF_AN    = 0,
  OFF_BN    = OFF_AN + N,
  OFF_R     = OFF_BN + M,
  OFF_CC    = OFF_R + N,
  OFF_R1    = OFF_CC + M,
  OFF_CC1   = OFF_R1 + N,
  OFF_RP    = OFF_CC1 + M,        // rowPrev
  OFF_CP    = OFF_RP + N,         // colPrev
  OFF_GPART = OFF_CP + M,         // 16 * N  (row-sum partials per col-chunk)
  OFF_HPART = OFF_GPART + 16 * N, // 16 * M  (col-sum partials per row-chunk)
  OFF_EPART = OFF_HPART + 16 * M, // 1024
  OFF_KPART = OFF_EPART + 1024,   // 1024
  OFF_APART = OFF_KPART + 1024,   // 8
  OFF_BPART = OFF_APART + 8,      // 8
  OFF_TPART = OFF_BPART + 8,      // 8
  OFF_SC    = OFF_TPART + 8       // scalars
};
enum { SC_S = 0, SC_SNEW = 1, SC_ERR = 2, SC_INVA = 3, SC_INVB = 4, SC_DONE = 5 };

__device__ inline float blockReduce256(float v, float* red) {
  const int t = threadIdx.x;
  red[t] = v;
  __syncthreads();
  for (int s = 128; s > 0; s >>= 1) {       // fixed tree -> deterministic
    if (t < s) red[t] += red[t + s];
    __syncthreads();
  }
  return red[0];
}

// ---------------------------- init kernels ---------------------------------

__global__ __launch_bounds__(256) void k_vec_init(float* R, float* Cc,
                                                  float* rp, float* cp) {
  const int i = blockIdx.x * 256 + threadIdx.x;   // 8 blocks -> 2048
  R[i] = 1.0f; Cc[i] = 1.0f; rp[i] = 1.0f; cp[i] = 1.0f;
}

__global__ __launch_bounds__(256) void k_sum_ab(const float* a, const float* b,
                                                float* apart, float* bpart) {
  __shared__ float red[256];
  const int part = blockIdx.x & 7;
  const bool isB = blockIdx.x >= 8;
  const float* src = isB ? b : a;
  const float s = blockReduce256(src[part * 256 + threadIdx.x], red);
  if (threadIdx.x == 0) (isB ? bpart : apart)[part] = s;
}

__global__ __launch_bounds__(256) void k_sumK0(const float* C, float* kpart) {
  __shared__ float red[256];
  const int base = blockIdx.x * 256 + threadIdx.x;
  float s = 0.0f;
  for (int it = 0; it < NM / STRIDE_ELEMS; ++it)
    s += __expf(NEG_INV_EPS * C[base + it * STRIDE_ELEMS]);
  const float tot = blockReduce256(s, red);
  if (threadIdx.x == 0) kpart[blockIdx.x] = tot;
}

__global__ __launch_bounds__(256) void k_scalar_init(const float* kpart,
                                                     const float* apart,
                                                     const float* bpart,
                                                     float* sc) {
  __shared__ float red[256];
  float s = 0.0f;
  for (int i = threadIdx.x; i < 1024; i += 256) s += kpart[i];
  const float sumK0 = blockReduce256(s, red);
  if (threadIdx.x == 0) {
    float sa = 0.0f, sb = 0.0f;
    for (int i = 0; i < 8; ++i) { sa += apart[i]; sb += bpart[i]; }
    sc[SC_S]    = S_MASS_C / sumK0;   // folds K0 *= m/sum(K0) into S
    sc[SC_INVA] = 1.0f / sa;
    sc[SC_INVB] = 1.0f / sb;
    sc[SC_ERR]  = 1.0f;
    ((int*)sc)[SC_DONE] = 0;
  }
}

__global__ __launch_bounds__(256) void k_normalize(const float* a, const float* b,
                                                   const float* sc,
                                                   float* an, float* bn) {
  const int idx = blockIdx.x * 256 + threadIdx.x;   // 16 blocks -> 4096
  if (idx < N) an[idx] = a[idx] * sc[SC_INVA];
  else         bn[idx - N] = b[idx - N] * sc[SC_INVB];
}

// ------------------- per-iteration: row GEMV via WMMA ----------------------
// g_i = sum_j exp(-10*C[i,j]) * Cc_j  over a 128-column chunk per wave.
// A tile 16x4 f32: lane<16 holds K=k,k+1; lane>=16 holds K=k+2,k+3 (row=lane&15).
// B tile 4x16 f32: VGPR v, half-wave h -> row K = v + 2h; columns replicated
// with Cc so every D column equals the partial row sum.

__global__ __launch_bounds__(32) void k_rowsum_wmma(const float* __restrict__ C,
                                                    const float* __restrict__ Cc,
                                                    float* __restrict__ gpart,
                                                    const float* __restrict__ sc) {
  if (((const int*)sc)[SC_DONE]) return;          // uniform: EXEC stays full
  const int r0 = blockIdx.x * 16;                 // 128 row tiles
  const int c0 = blockIdx.y * 128;                // 16 column chunks
  const int lane = threadIdx.x;
  const int half = lane >> 4, mrow = lane & 15;
  const float* rowp = C + (size_t)(r0 + mrow) * M + c0;
  const float* cvp  = Cc + c0;

  v8f acc = {};
  for (int k = 0; k < 128; k += 4) {
    v2f a, b;
    const int ko = k + 2 * half;
    a.x = __expf(NEG_INV_EPS * rowp[ko]);         // 16B-contiguous per row
    a.y = __expf(NEG_INV_EPS * rowp[ko + 1]);
    b.x = cvp[ko];                                // B row K=v+2h, replicated
    b.y = cvp[ko + 1];
    acc = __builtin_amdgcn_wmma_f32_16x16x4_f32(
        /*neg_a=*/false, a, /*neg_b=*/false, b,
        /*c_mod=*/(short)0, acc, /*reuse_a=*/false, /*reuse_b=*/false);
  }
  // D layout: VGPR v -> row v (lanes 0-15) / row v+8 (lanes 16-31)
  if (lane == 0) {
    for (int v = 0; v < 8; ++v) gpart[blockIdx.y * N + r0 + v] = acc[v];
  } else if (lane == 16) {
    for (int v = 0; v < 8; ++v) gpart[blockIdx.y * N + r0 + 8 + v] = acc[v];
  }
}

__global__ __launch_bounds__(256) void k_row_update(const float* gpart,
                                                    const float* an,
                                                    float* rowPrev,
                                                    const float* R, float* R1,
                                                    const float* sc) {
  if (((const int*)sc)[SC_DONE]) return;
  const int i = blockIdx.x * 256 + threadIdx.x;
  float g = 0.0f;
#pragma unroll
  for (int ch = 0; ch < 16; ++ch) g += gpart[ch * N + i];
  const float S  = sc[SC_S];
  const float rv = fminf(an[i] * rowPrev[i] / (R[i] * S * g), 1.0f);
  R1[i] = R[i] * rv / rowPrev[i];
  rowPrev[i] = rv;
}

// h_j = sum_i exp(-10*C[i,j]) * R1_i  (coalesced over columns, rows staged)
__global__ __launch_bounds__(128) void k_colsum(const float* __restrict__ C,
                                                const float* __restrict__ R1,
                                                float* __restrict__ hpart,
                                                const float* __restrict__ sc) {
  if (((const int*)sc)[SC_DONE]) return;
  __shared__ float rv[128];
  const int j  = blockIdx.x * 128 + threadIdx.x;  // 16 col chunks
  const int r0 = blockIdx.y * 128;                // 16 row chunks
  rv[threadIdx.x] = R1[r0 + threadIdx.x];
  __syncthreads();
  float s = 0.0f;
  for (int r = 0; r < 128; ++r)
    s += __expf(NEG_INV_EPS * C[(size_t)(r0 + r) * M + j]) * rv[r];
  hpart[blockIdx.y * M + j] = s;
}

__global__ __launch_bounds__(256) void k_col_update(const float* hpart,
                                                    const float* bn,
                                                    float* colPrev,
                                                    const float* Cc, float* Cc1,
                                                    float* Tpart,
                                                    const float* sc) {
  __shared__ float red[256];
  if (((const int*)sc)[SC_DONE]) return;          // uniform for whole grid
  const int j = blockIdx.x * 256 + threadIdx.x;
  float h = 0.0f;
#pragma unroll
  for (int ch = 0; ch < 16; ++ch) h += hpart[ch * M + j];
  const float S   = sc[SC_S];
  const float cv  = fminf(bn[j] * colPrev[j] / (Cc[j] * S * h), 1.0f);
  const float cc1 = Cc[j] * cv / colPrev[j];
  Cc1[j] = cc1;
  colPrev[j] = cv;
  const float t = blockReduce256(h * cc1, red);   // partial of T = sum h*Cc1
  if (threadIdx.x == 0) Tpart[blockIdx.x] = t;
}

__global__ __launch_bounds__(32) void k_tscalar(const float* Tpart, float* sc) {
  if (threadIdx.x != 0) return;
  if (((const int*)sc)[SC_DONE]) return;
  float T = 0.0f;
  for (int i = 0; i < 8; ++i) T += Tpart[i];      // sum(K2) = S*T
  sc[SC_SNEW] = S_MASS_C / T;                     // S' = S*m/sum(K2) = m/T
}

// err^2 = sum K0^2 * (S*R_i*Cc_j - S'*R1_i*Cc1_j)^2  (old vs new K)
__global__ __launch_bounds__(256) void k_err_partial(const float* __restrict__ C,
                                                     const float* R, const float* Cc,
                                                     const float* R1, const float* Cc1,
                                                     const float* sc,
                                                     float* epart) {
  __shared__ float red[256];
  const float S = sc[SC_S], Sn = sc[SC_SNEW];
  const int base = blockIdx.x * 256 + threadIdx.x;
  float acc = 0.0f;
  for (int it = 0; it < NM / STRIDE_ELEMS; ++it) {
    const int idx = base + it * STRIDE_ELEMS;
    if (it + 1 < NM / STRIDE_ELEMS)
      __builtin_prefetch(C + idx + STRIDE_ELEMS, 0, 0);  // global_prefetch_b8
    const int r = idx >> 11, c = idx & (M - 1);
    const float k0 = __expf(NEG_INV_EPS * C[idx]);
    const float d  = k0 * (S * R[r] * Cc[c] - Sn * R1[r] * Cc1[c]);
    acc += d * d;
  }
  const float tot = blockReduce256(acc, red);
  if (threadIdx.x == 0) epart[blockIdx.x] = tot;
}

__global__ __launch_bounds__(256) void k_commit(float* R, const float* R1,
                                                float* Cc, const float* Cc1,
                                                float* sc) {
  if (((const int*)sc)[SC_DONE]) return;          // freeze once converged
  const int i = blockIdx.x * 256 + threadIdx.x;
  if (i < N)          R[i] = R1[i];
  else if (i < N + M) Cc[i - N] = Cc1[i - N];
  else if (i == N + M) sc[SC_S] = sc[SC_SNEW];
}

__global__ __launch_bounds__(256) void k_err_final(const float* epart, float* sc) {
  __shared__ float red[256];
  float s = 0.0f;
  for (int i = threadIdx.x; i < 1024; i += 256) s += epart[i];
  const float tot = blockReduce256(s, red);
  if (threadIdx.x == 0) {
    int* isc = (int*)sc;
    if (!isc[SC_DONE]) {
      const float err = sqrtf(tot);
      sc[SC_ERR] = err;
      if (err <= TOL_C) isc[SC_DONE] = 1;         // done set at end of iter
    }
  }
}

__global__ __launch_bounds__(256) void k_final(const float* __restrict__ C,
                                               const float* R, const float* Cc,
                                               const float* sc,
                                               float* __restrict__ out) {
  const int idx = blockIdx.x * 256 + threadIdx.x;  // 16384 blocks
  const int r = idx >> 11, c = idx & (M - 1);
  out[idx] = __expf(NEG_INV_EPS * C[idx]) * R[r] * Cc[c] * sc[SC_S];
}

// ---------------------------------------------------------------------------

extern "C" void kernel_launch(void* const* d_in, const int* in_sizes, int n_in,
                              void* d_out, int out_size, void* d_ws, size_t ws_size,
                              hipStream_t stream) {
  const float* C = (const float*)d_in[0];
  const float* a = (const float*)d_in[1];
  const float* b = (const float*)d_in[2];
  float* out = (float*)d_out;
  float* w   = (float*)d_ws;

  float *an = w + OFF_AN,  *bn = w + OFF_BN;
  float *R  = w + OFF_R,   *Cc = w + OFF_CC;
  float *R1 = w + OFF_R1,  *Cc1 = w + OFF_CC1;
  float *rp = w + OFF_RP,  *cp = w + OFF_CP;
  float *gpart = w + OFF_GPART, *hpart = w + OFF_HPART;
  float *epart = w + OFF_EPART, *kpart = w + OFF_KPART;
  float *apart = w + OFF_APART, *bpart = w + OFF_BPART;
  float *Tpart = w + OFF_TPART, *sc    = w + OFF_SC;

  // init: normalize a,b; S = m / sum(exp(-C/eps)); R=Cc=prev=1; err=1; done=0
  k_vec_init   <<<8,    256, 0, stream>>>(R, Cc, rp, cp);
  k_sum_ab     <<<16,   256, 0, stream>>>(a, b, apart, bpart);
  k_sumK0      <<<1024, 256, 0, stream>>>(C, kpart);
  k_scalar_init<<<1,    256, 0, stream>>>(kpart, apart, bpart, sc);
  k_normalize  <<<16,   256, 0, stream>>>(a, b, sc, an, bn);

  for (int t = 0; t < N_ITERS_C; ++t) {
    k_rowsum_wmma<<<dim3(128, 16), 32,  0, stream>>>(C, Cc, gpart, sc);
    k_row_update <<<8,             256, 0, stream>>>(gpart, an, rp, R, R1, sc);
    k_colsum     <<<dim3(16, 16),  128, 0, stream>>>(C, R1, hpart, sc);
    k_col_update <<<8,             256, 0, stream>>>(hpart, bn, cp, Cc, Cc1, Tpart, sc);
    k_tscalar    <<<1,             32,  0, stream>>>(Tpart, sc);
    if (t % 10 == 0)   // cpt%10==0 path (cpt==t while not done; frozen after)
      k_err_partial<<<1024, 256, 0, stream>>>(C, R, Cc, R1, Cc1, sc, epart);
    k_commit     <<<17, 256, 0, stream>>>(R, R1, Cc, Cc1, sc);
    if (t % 10 == 0)
      k_err_final<<<1, 256, 0, stream>>>(epart, sc);
  }

  k_final<<<NM / 256, 256, 0, stream>>>(C, R, Cc, sc, out);

  (void)in_sizes; (void)n_in; (void)out_size; (void)ws_size;
}